// Dynamic_relu_b_62242666054404
// MI455X (gfx1250) — compile-verified
//
#include <hip/hip_runtime.h>

typedef __attribute__((ext_vector_type(2))) float v2f;
typedef __attribute__((ext_vector_type(4))) float v4f;
typedef __attribute__((ext_vector_type(8))) float v8f;

#define BS   32
#define CH   256
#define HW   4096
#define HID  32
#define NOUT 1024   // K * 2C = 2 * 512
#define PSTR 258    // pooled LDS row stride (pad: conflict-free A-frag reads)
#define HSTR 34     // h LDS row stride (even pad: aligned b64 + conflict-free)

// ---------------- Kernel 1: global average pool over H*W ----------------
// One 256-thread block per (b,c) pair; 4096 floats = 1024 float4, coalesced.
__global__ __launch_bounds__(256) void pool_kernel(const float* __restrict__ x,
                                                   float* __restrict__ pooled) {
  const int bc = blockIdx.x;  // b*256 + c
  const v4f* __restrict__ x4 = (const v4f*)x + (size_t)bc * (HW / 4);
  float s = 0.f;
  for (int i = threadIdx.x; i < HW / 4; i += 256) {
    v4f v = x4[i];
    s += (v.x + v.y) + (v.z + v.w);
  }
  // wave32 butterfly reduce
  for (int off = 16; off > 0; off >>= 1) s += __shfl_xor(s, off, 32);
  __shared__ float wsum[8];
  const int lane = threadIdx.x & 31, wv = threadIdx.x >> 5;
  if (lane == 0) wsum[wv] = s;
  __syncthreads();
  if (threadIdx.x == 0) {
    float t = 0.f;
#pragma unroll
    for (int i = 0; i < 8; ++i) t += wsum[i];
    pooled[bc] = t * (1.0f / HW);
  }
}

// ---------------- Kernel 2: MLP -> per-(b,c) affine coefficients ----------------
// Single workgroup (8 waves). GEMM1 (32x32,K=256) via f32 WMMA on waves 0-3,
// GEMM2 (32x1024,K=32) via f32 WMMA on all 8 waves. coef[b][c] = {a0,b0,a1,b1}.
__global__ __launch_bounds__(256) void coef_kernel(
    const float* __restrict__ pooled, const float* __restrict__ w1,
    const float* __restrict__ b1, const float* __restrict__ w2,
    const float* __restrict__ b2, float* __restrict__ coef) {
  __shared__ float plds[BS * PSTR];   // pooled, padded
  __shared__ float hlds[BS * HSTR];   // hidden activations, padded
  const int tid  = threadIdx.x;
  const int lane = tid & 31;
  const int wv   = tid >> 5;

  // Stage pooled (32x256) into LDS, coalesced.
  for (int i = tid; i < BS * CH; i += 256)
    plds[(i >> 8) * PSTR + (i & 255)] = pooled[i];
  __syncthreads();

  // GEMM1: h = relu(pooled @ w1^T + b1). 2x2 tiles of 16x16; wave w -> tile.
  if (wv < 4) {
    const int mi = wv >> 1, ni = wv & 1;
    const int mrow = mi * 16 + (lane & 15);
    const int ncol = ni * 16 + (lane & 15);
    const int ksel = (lane >> 4) << 1;  // lanes 16-31 handle K+2,K+3
    v8f acc = {};
    for (int k = 0; k < CH; k += 4) {
      const int kb = k + ksel;
      v2f a, b;
      a.x = plds[mrow * PSTR + kb];
      a.y = plds[mrow * PSTR + kb + 1];
      b.x = w1[ncol * CH + kb];        // B[k][n] = w1[n][k]
      b.y = w1[ncol * CH + kb + 1];
      acc = __builtin_amdgcn_wmma_f32_16x16x4_f32(false, a, false, b,
                                                  (short)0, acc, false, false);
    }
    const float bias  = b1[ncol];
    const int   mbase = mi * 16 + ((lane < 16) ? 0 : 8);
#pragma unroll
    for (int r = 0; r < 8; ++r) {
      const float h = acc[r] + bias;
      hlds[(mbase + r) * HSTR + ncol] = fmaxf(h, 0.f);
    }
  }
  __syncthreads();

  // GEMM2: z = h @ w2'^T + b2 (32x1024, K=32). 128 tiles, 16 per wave.
  for (int t = wv * 16; t < wv * 16 + 16; ++t) {
    const int mi = t >> 6;                    // 0..1
    const int nj = t & 63;                    // 0..63
    const int mrow = mi * 16 + (lane & 15);
    const int ncol = nj * 16 + (lane & 15);   // o' in 0..1023
    const int ksel = (lane >> 4) << 1;
    v8f acc = {};
#pragma unroll
    for (int k = 0; k < HID; k += 4) {
      const int kb = k + ksel;
      v2f a, b;
      a.x = hlds[mrow * HSTR + kb];
      a.y = hlds[mrow * HSTR + kb + 1];
      b.x = w2[ncol * HID + kb];              // B[k][n] = w2flat[n][k]
      b.y = w2[ncol * HID + kb + 1];
      acc = __builtin_amdgcn_wmma_f32_16x16x4_f32(false, a, false, b,
                                                  (short)0, acc, false, false);
    }
    const float bias  = b2[ncol];
    const int   kidx  = ncol >> 9;            // which of K=2 experts
    const int   o     = ncol & 511;
    const int   c     = o >> 1;               // channel
    const int   isB   = o & 1;                // even->alpha, odd->beta
    const float lam   = isB ? 0.5f : 1.0f;    // LAMBDA_BETA / LAMBDA_ALPHA
    const float init  = (kidx == 0) ? 1.0f : 0.0f;
    const int   mbase = mi * 16 + ((lane < 16) ? 0 : 8);
#pragma unroll
    for (int r = 0; r < 8; ++r) {
      const float z     = acc[r] + bias;
      const float delta = tanhf(0.5f * z);    // == 2*sigmoid(z) - 1
      coef[((mbase + r) * CH + c) * 4 + kidx * 2 + isB] = init + lam * delta;
    }
  }
}

// ---------------- Kernel 3: elementwise max of two affines ----------------
// out = max(a0*x+b0, a1*x+b1). NT stores so x stays L2-resident during pass.
__global__ __launch_bounds__(256) void apply_kernel(const float* __restrict__ x,
                                                    const v4f* __restrict__ coef,
                                                    float* __restrict__ out) {
  const int bc = blockIdx.x;
  const v4f cf = coef[bc];  // {a0, b0, a1, b1}
  const v4f* __restrict__ x4 = (const v4f*)x + (size_t)bc * (HW / 4);
  v4f* __restrict__ o4 = (v4f*)out + (size_t)bc * (HW / 4);
  for (int i = threadIdx.x; i < HW / 4; i += 256) {
    const v4f v = x4[i];
    v4f r;
    r.x = fmaxf(fmaf(v.x, cf.x, cf.y), fmaf(v.x, cf.z, cf.w));
    r.y = fmaxf(fmaf(v.y, cf.x, cf.y), fmaf(v.y, cf.z, cf.w));
    r.z = fmaxf(fmaf(v.z, cf.x, cf.y), fmaf(v.z, cf.z, cf.w));
    r.w = fmaxf(fmaf(v.w, cf.x, cf.y), fmaf(v.w, cf.z, cf.w));
    __builtin_nontemporal_store(r, &o4[i]);
  }
}

extern "C" void kernel_launch(void* const* d_in, const int* in_sizes, int n_in,
                              void* d_out, int out_size, void* d_ws, size_t ws_size,
                              hipStream_t stream) {
  const float* x  = (const float*)d_in[0];
  const float* w1 = (const float*)d_in[1];
  const float* b1 = (const float*)d_in[2];
  const float* w2 = (const float*)d_in[3];
  const float* b2 = (const float*)d_in[4];
  float* out = (float*)d_out;

  float* pooled = (float*)d_ws;        // 32*256 floats   (32 KiB)
  float* coef   = pooled + BS * CH;    // 32*256*4 floats (128 KiB)

  pool_kernel<<<BS * CH, 256, 0, stream>>>(x, pooled);
  coef_kernel<<<1, 256, 0, stream>>>(pooled, w1, b1, w2, b2, coef);
  apply_kernel<<<BS * CH, 256, 0, stream>>>(x, coef ? (const v4f*)coef : nullptr, out);
}